// GCN_83545703842213
// MI455X (gfx1250) — compile-verified
//
#include <hip/hip_runtime.h>
#include <hip/hip_bf16.h>

typedef float v2f __attribute__((ext_vector_type(2)));
typedef float v8f __attribute__((ext_vector_type(8)));

#define HDIM 128
#define LDS_STRIDE 130   // pad 128->130: LDS bank = (2*m + k) % 64, conflict-free

// ---------------------------------------------------------------------------
// deg[n] = 1 (self loop weight), then atomic accumulation of edge weights
// ---------------------------------------------------------------------------
__global__ __launch_bounds__(256) void k_init_deg(float* __restrict__ deg, int N) {
    int i = blockIdx.x * blockDim.x + threadIdx.x;
    if (i < N) deg[i] = 1.0f;
}

__global__ __launch_bounds__(256) void k_deg_accum(const int* __restrict__ dst,
                                                   const float* __restrict__ ew,
                                                   float* __restrict__ deg, int E) {
    int i = blockIdx.x * blockDim.x + threadIdx.x;
    if (i < E) unsafeAtomicAdd(&deg[dst[i]], ew[i]);   // global_atomic_add_f32 @ L2
}

__global__ __launch_bounds__(256) void k_dinv(float* __restrict__ deg, int N) {
    int i = blockIdx.x * blockDim.x + threadIdx.x;
    if (i < N) {
        float d = deg[i];
        deg[i] = (d > 0.0f) ? rsqrtf(d) : 0.0f;
    }
}

// ---------------------------------------------------------------------------
// out[16r : 16r+16, :] = A[16r : 16r+16, :] @ W  using V_WMMA_F32_16X16X4_F32
// 256 threads = 8 waves; wave w owns columns [16w, 16w+16).
// A tile staged in LDS (shared by the 8 waves), optional fused ReLU.
// ---------------------------------------------------------------------------
__global__ __launch_bounds__(256) void k_gemm_wmma(const float* __restrict__ A,
                                                   const float* __restrict__ W,
                                                   float* __restrict__ out,
                                                   int apply_relu) {
    __shared__ float a_lds[16 * LDS_STRIDE];

    const int tid      = threadIdx.x;
    const int row_base = blockIdx.x * 16;

    // Cooperative stage of the 16x128 A tile (8 floats / thread), fused ReLU.
    #pragma unroll
    for (int i = 0; i < 8; ++i) {
        int idx = tid * 8 + i;            // 0..2047
        int r = idx >> 7, c = idx & 127;
        float v = A[(size_t)(row_base + r) * HDIM + c];
        if (apply_relu) v = fmaxf(v, 0.0f);
        a_lds[r * LDS_STRIDE + c] = v;
    }
    __syncthreads();

    const int wave     = tid >> 5;        // 0..7
    const int lane     = tid & 31;
    const int col_base = wave * 16;
    const int m        = lane & 15;       // A row (lanes 0-15) / same for hi half
    const int kk       = (lane >> 4) << 1;// K subpair: 0 (lanes 0-15) or 2 (16-31)

    v8f acc = {};
    const float* wp = W + col_base + m;   // B column n == lane&15

    #pragma unroll
    for (int k = 0; k < HDIM; k += 4) {
        v2f a, b;
        // ISA 7.12.2: 32-bit A 16x4 -> lane m holds K=kk in v0, K=kk+1 in v1
        a.x = a_lds[m * LDS_STRIDE + k + kk];
        a.y = a_lds[m * LDS_STRIDE + k + kk + 1];
        // B 4x16: lane n holds K=kk in v0, K=kk+1 in v1
        b.x = wp[(size_t)(k + kk) * HDIM];
        b.y = wp[(size_t)(k + kk + 1) * HDIM];
        acc = __builtin_amdgcn_wmma_f32_16x16x4_f32(false, a, false, b,
                                                    (short)0, acc, false, false);
    }

    // C/D layout: VGPR i -> M=i (lanes 0-15) / M=8+i (lanes 16-31), N=lane&15
    const int n     = lane & 15;
    const int rhalf = (lane >> 4) * 8;
    #pragma unroll
    for (int i = 0; i < 8; ++i)
        out[(size_t)(row_base + rhalf + i) * HDIM + col_base + n] = acc[i];
}

// ---------------------------------------------------------------------------
// h[n, j] = bias[j] + dinv[n]^2 * xw[n, j]   (self-loop term + bias)
// ---------------------------------------------------------------------------
__global__ __launch_bounds__(256) void k_init_out(const float* __restrict__ xw,
                                                  const float* __restrict__ dinv,
                                                  const float* __restrict__ bias,
                                                  float* __restrict__ h, int total) {
    int i = blockIdx.x * blockDim.x + threadIdx.x;
    if (i < total) {
        int n = i >> 7, j = i & 127;
        float di = dinv[n];
        h[i] = bias[j] + di * di * xw[i];
    }
}

// ---------------------------------------------------------------------------
// One wave32 per edge: lane owns a float4 slice of the 128-wide feature row.
// gather xw[src] (b128), scale, scatter-add into h[dst] via L2 f32 atomics.
// ---------------------------------------------------------------------------
__global__ __launch_bounds__(256) void k_edge_scatter(const int* __restrict__ src,
                                                      const int* __restrict__ dst,
                                                      const float* __restrict__ ew,
                                                      const float* __restrict__ dinv,
                                                      const float* __restrict__ xw,
                                                      float* __restrict__ h, int E) {
    int e    = (blockIdx.x * blockDim.x + threadIdx.x) >> 5;
    int lane = threadIdx.x & 31;
    if (e >= E) return;

    int   s    = src[e];
    int   d    = dst[e];
    float norm = dinv[s] * ew[e] * dinv[d];

    const float4* xp = (const float4*)(xw + (size_t)s * HDIM) + lane;
    float4 v = *xp;

    float* op = h + (size_t)d * HDIM + lane * 4;
    unsafeAtomicAdd(op + 0, v.x * norm);
    unsafeAtomicAdd(op + 1, v.y * norm);
    unsafeAtomicAdd(op + 2, v.z * norm);
    unsafeAtomicAdd(op + 3, v.w * norm);
}

// ---------------------------------------------------------------------------
// Row L2-normalize in place: one wave per node, shfl_xor reduction.
// ---------------------------------------------------------------------------
__global__ __launch_bounds__(256) void k_row_normalize(float* __restrict__ h, int N) {
    int n    = (blockIdx.x * blockDim.x + threadIdx.x) >> 5;
    int lane = threadIdx.x & 31;
    if (n >= N) return;

    float4* p = (float4*)(h + (size_t)n * HDIM) + lane;
    float4 v = *p;
    float s = v.x * v.x + v.y * v.y + v.z * v.z + v.w * v.w;
    #pragma unroll
    for (int off = 16; off > 0; off >>= 1)
        s += __shfl_xor(s, off, 32);
    float scale = 1.0f / fmaxf(sqrtf(s), 1e-12f);
    v.x *= scale; v.y *= scale; v.z *= scale; v.w *= scale;
    *p = v;
}

// ---------------------------------------------------------------------------
extern "C" void kernel_launch(void* const* d_in, const int* in_sizes, int n_in,
                              void* d_out, int out_size, void* d_ws, size_t ws_size,
                              hipStream_t stream) {
    const float* x   = (const float*)d_in[0];            // [N,128]
    const int*   ei  = (const int*)  d_in[1];            // [2,E]
    const float* ew  = (const float*)d_in[2];            // [E]
    const float* W1  = (const float*)d_in[3];            // [128,128]
    const float* b1  = (const float*)d_in[4];            // [128]
    const float* W2  = (const float*)d_in[5];            // [128,128]
    const float* b2  = (const float*)d_in[6];            // [128]

    const int N = in_sizes[0] / HDIM;                    // 100000 (divisible by 16)
    const int E = in_sizes[2];                           // 1600000
    const int* src = ei;
    const int* dstp = ei + E;

    // Workspace layout: dinv[N] | xw[N*128] | h1[N*128]
    char* ws = (char*)d_ws;
    size_t off = ((size_t)N * sizeof(float) + 255) & ~(size_t)255;
    float* dinv = (float*)ws;
    float* xw   = (float*)(ws + off);
    float* h1   = (float*)(ws + off + (size_t)N * HDIM * sizeof(float));
    float* h2   = (float*)d_out;

    const int total   = N * HDIM;
    const int blkN    = (N + 255) / 256;
    const int blkE    = (E + 255) / 256;
    const int blkTot  = (total + 255) / 256;
    const int blkEdgW = (int)(((size_t)E * 32 + 255) / 256);  // 1 wave per edge
    const int blkRowW = (int)(((size_t)N * 32 + 255) / 256);  // 1 wave per node
    const int blkGemm = N / 16;                               // 6250 tiles of rows

    // Degree / normalization coefficients
    k_init_deg <<<blkN, 256, 0, stream>>>(dinv, N);
    k_deg_accum<<<blkE, 256, 0, stream>>>(dstp, ew, dinv, E);
    k_dinv     <<<blkN, 256, 0, stream>>>(dinv, N);

    // Layer 1: xw = x @ W1 ; h1 = b1 + dinv^2*xw ; scatter edges
    k_gemm_wmma   <<<blkGemm, 256, 0, stream>>>(x, W1, xw, 0);
    k_init_out    <<<blkTot,  256, 0, stream>>>(xw, dinv, b1, h1, total);
    k_edge_scatter<<<blkEdgW, 256, 0, stream>>>(src, dstp, ew, dinv, xw, h1, E);

    // Layer 2: xw = relu(h1) @ W2 (ReLU fused into A-tile stage); accumulate in d_out
    k_gemm_wmma   <<<blkGemm, 256, 0, stream>>>(h1, W2, xw, 1);
    k_init_out    <<<blkTot,  256, 0, stream>>>(xw, dinv, b2, h2, total);
    k_edge_scatter<<<blkEdgW, 256, 0, stream>>>(src, dstp, ew, dinv, xw, h2, E);

    // Row-wise L2 normalization in place in d_out
    k_row_normalize<<<blkRowW, 256, 0, stream>>>(h2, N);
}